// DSGCNLayer_10823317586648
// MI455X (gfx1250) — compile-verified
//
#include <hip/hip_runtime.h>
#include <hip/hip_bf16.h>

// Problem constants (match reference)
#define B_  8
#define T_  512
#define N_  47
#define CIN 128
#define COUT 256
#define K_  3
#define OP_ 512           // COUT (h) + COUT (residual) concatenated
#define EPS_ 1e-5f

typedef __attribute__((ext_vector_type(2))) float v2f;
typedef __attribute__((ext_vector_type(8))) float v8f;

// ---------------------------------------------------------------------------
// Pass 0a: s[k,n] = sum_m A_eff / max(sum_m |A_eff|, 1)
// ---------------------------------------------------------------------------
__global__ void pass0_s(const float* __restrict__ A,
                        const float* __restrict__ adj,
                        float* __restrict__ s_ws) {
    int id = blockIdx.x * blockDim.x + threadIdx.x;
    if (id < K_ * N_) {
        int k = id / N_, n = id % N_;
        const float* ar = A   + (size_t)(k * N_ + n) * N_;
        const float* jr = adj + (size_t)(k * N_ + n) * N_;
        float sum = 0.f, asum = 0.f;
        for (int m = 0; m < N_; ++m) {
            float a = ar[m] + tanhf(jr[m]) * 0.3f;
            sum += a; asum += fabsf(a);
        }
        s_ws[id] = sum / fmaxf(asum, 1.0f);
    }
}

// ---------------------------------------------------------------------------
// Pass 0b: W_cat[n][o'][c]; o'<256 -> effective pointwise weight, else W_res
// ---------------------------------------------------------------------------
__global__ __launch_bounds__(CIN) void pass0_wcat(const float* __restrict__ Wpw,
                                                  const float* __restrict__ Wres,
                                                  const float* __restrict__ dw,
                                                  const float* __restrict__ s_ws,
                                                  float* __restrict__ Wcat) {
    int blk = blockIdx.x;             // 47*512 blocks
    int n   = blk >> 9;               // / 512
    int op  = blk & (OP_ - 1);
    int c   = threadIdx.x;            // 0..127
    float w;
    if (op < COUT) {
        w = 0.f;
        #pragma unroll
        for (int k = 0; k < K_; ++k)
            w += s_ws[k * N_ + n] * dw[k * CIN + c] * Wpw[(size_t)op * (K_ * CIN) + k * CIN + c];
    } else {
        w = Wres[(size_t)(op - COUT) * CIN + c];
    }
    Wcat[((size_t)n * OP_ + op) * CIN + c] = w;
}

// ---------------------------------------------------------------------------
// Pass 1: per (b,n,t-block) GEMM via V_WMMA_F32_16X16X4_F32
//   hres[b][n][t][o'] = sum_c x[b][t][n][c] * W_cat[n][o'][c]
// ---------------------------------------------------------------------------
__global__ __launch_bounds__(256) void pass1_gemm(const float* __restrict__ x,
                                                  const float* __restrict__ Wcat,
                                                  float* __restrict__ hres) {
    const int tb = blockIdx.x;        // 0..3  (128 t each)
    const int n  = blockIdx.y;        // 0..46
    const int b  = blockIdx.z;        // 0..7
    const int t0 = tb * 128;

    __shared__ float xs[128][128];    // 64 KB x tile [t_local][c]

    // cooperative coalesced load: x[b, t0..t0+127, n, 0..127]
    const float* xbase = x + (((size_t)b * T_ + t0) * N_ + n) * CIN;
    for (int idx = threadIdx.x; idx < 128 * 32; idx += 256) {
        const int row = idx >> 5;
        const int c4  = (idx & 31) << 2;
        const float4 v = *(const float4*)(xbase + (size_t)row * (N_ * CIN) + c4);
        xs[row][c4 + 0] = v.x; xs[row][c4 + 1] = v.y;
        xs[row][c4 + 2] = v.z; xs[row][c4 + 3] = v.w;
    }
    __syncthreads();

    const int lane = threadIdx.x & 31;
    const int wave = threadIdx.x >> 5;          // 0..7, each wave: 16 t rows
    const int mrow = lane & 15;
    const int koff = (lane >> 4) * 2;           // ISA A/B fragment layout

    // A strip: 16 t x 128 c, kept in 64 VGPRs (32 x v2f)
    v2f a[32];
    const int tl = wave * 16 + mrow;
    #pragma unroll
    for (int s = 0; s < 32; ++s) {
        a[s].x = xs[tl][4 * s + koff + 0];
        a[s].y = xs[tl][4 * s + koff + 1];
    }

    const float* wn = Wcat + (size_t)n * OP_ * CIN;
    float* outbase = hres + (((size_t)b * N_ + n) * T_ + t0 + wave * 16) * OP_;
    const int trow_off = (lane >> 4) * 8;       // ISA C/D layout

    for (int ot = 0; ot < 32; ++ot) {           // 32 o'-tiles of 16
        const int ob = ot * 16;
        const float* wrow = wn + (size_t)(ob + mrow) * CIN + koff;
        v8f acc = {0.f, 0.f, 0.f, 0.f, 0.f, 0.f, 0.f, 0.f};
        #pragma unroll
        for (int s = 0; s < 32; ++s) {          // K = 128 in steps of 4
            v2f bf;
            bf.x = wrow[4 * s + 0];
            bf.y = wrow[4 * s + 1];
            acc = __builtin_amdgcn_wmma_f32_16x16x4_f32(
                false, a[s], false, bf, (short)0, acc, false, false);
        }
        const int oc = ob + mrow;
        #pragma unroll
        for (int j = 0; j < 8; ++j)
            outbase[(size_t)(j + trow_off) * OP_ + oc] = acc[j];
    }
}

// ---------------------------------------------------------------------------
// Pass 2: depthwise conv3 over T + GroupNorm(8) + residual + LayerNorm + GELU
// WG per (b,n); 16 waves; wave w owns t in [32w, 32w+32); lane holds channels
// c = lane + 32*i (i = GroupNorm group index).
// ---------------------------------------------------------------------------
__global__ __launch_bounds__(512) void pass2_fuse(const float* __restrict__ hres,
                                                  const float* __restrict__ Wconv,
                                                  const float* __restrict__ gng,
                                                  const float* __restrict__ gnb,
                                                  const float* __restrict__ lng,
                                                  const float* __restrict__ lnb,
                                                  float* __restrict__ out) {
    const int bn = blockIdx.x;
    const int b = bn / N_, n = bn % N_;
    const int lane = threadIdx.x & 31;
    const int wave = threadIdx.x >> 5;   // 0..15

    __shared__ float partS[16][8];
    __shared__ float partQ[16][8];
    __shared__ float gmean[8];
    __shared__ float grstd[8];

    const float* hb = hres + ((size_t)b * N_ + n) * T_ * OP_;   // [t][512]

    // per-lane channel constants
    float wc0[8], wc1[8], wc2[8], gg[8], gb[8], lg[8], lb[8];
    #pragma unroll
    for (int i = 0; i < 8; ++i) {
        int c = lane + 32 * i;
        wc0[i] = Wconv[c * 3 + 0]; wc1[i] = Wconv[c * 3 + 1]; wc2[i] = Wconv[c * 3 + 2];
        gg[i] = gng[c]; gb[i] = gnb[c];
        lg[i] = lng[c]; lb[i] = lnb[c];
    }

    const int tstart = wave * 32;

    // ---- Phase A: conv stats per group (rolling window) ----
    float prev[8], cur[8], nxt[8];
    float lsum[8], lsq[8];
    #pragma unroll
    for (int i = 0; i < 8; ++i) {
        int c = lane + 32 * i;
        prev[i] = (tstart > 0) ? hb[(size_t)(tstart - 1) * OP_ + c] : 0.f;
        cur[i]  = hb[(size_t)tstart * OP_ + c];
        lsum[i] = 0.f; lsq[i] = 0.f;
    }
    for (int t = tstart; t < tstart + 32; ++t) {
        const bool hasnext = (t + 1) < T_;
        #pragma unroll
        for (int i = 0; i < 8; ++i) {
            int c = lane + 32 * i;
            nxt[i] = hasnext ? hb[(size_t)(t + 1) * OP_ + c] : 0.f;
            float cv = wc0[i] * prev[i] + wc1[i] * cur[i] + wc2[i] * nxt[i];
            lsum[i] += cv; lsq[i] += cv * cv;
            prev[i] = cur[i]; cur[i] = nxt[i];
        }
    }
    #pragma unroll
    for (int i = 0; i < 8; ++i) {
        float s = lsum[i], q = lsq[i];
        for (int off = 16; off > 0; off >>= 1) {
            s += __shfl_xor(s, off, 32);
            q += __shfl_xor(q, off, 32);
        }
        if (lane == 0) { partS[wave][i] = s; partQ[wave][i] = q; }
    }
    __syncthreads();
    if (threadIdx.x < 8) {          // deterministic fixed-order reduction
        float s = 0.f, q = 0.f;
        for (int w = 0; w < 16; ++w) { s += partS[w][threadIdx.x]; q += partQ[w][threadIdx.x]; }
        const float inv = 1.0f / (32.0f * (float)T_);
        float m = s * inv;
        float v = q * inv - m * m;
        gmean[threadIdx.x] = m;
        grstd[threadIdx.x] = rsqrtf(v + EPS_);
    }
    __syncthreads();

    float mean_[8], rs_[8];
    #pragma unroll
    for (int i = 0; i < 8; ++i) { mean_[i] = gmean[i]; rs_[i] = grstd[i]; }

    // ---- Phase B: conv + GN + residual + LN + GELU ----
    #pragma unroll
    for (int i = 0; i < 8; ++i) {
        int c = lane + 32 * i;
        prev[i] = (tstart > 0) ? hb[(size_t)(tstart - 1) * OP_ + c] : 0.f;
        cur[i]  = hb[(size_t)tstart * OP_ + c];
    }
    for (int t = tstart; t < tstart + 32; ++t) {
        const bool hasnext = (t + 1) < T_;
        float y[8];
        float ls = 0.f, lq = 0.f;
        #pragma unroll
        for (int i = 0; i < 8; ++i) {
            int c = lane + 32 * i;
            nxt[i] = hasnext ? hb[(size_t)(t + 1) * OP_ + c] : 0.f;
            float cv = wc0[i] * prev[i] + wc1[i] * cur[i] + wc2[i] * nxt[i];
            prev[i] = cur[i]; cur[i] = nxt[i];
            float g = (cv - mean_[i]) * rs_[i] * gg[i] + gb[i];
            float r = hb[(size_t)t * OP_ + COUT + c];     // residual half
            float yy = g + r;
            y[i] = yy; ls += yy; lq += yy * yy;
        }
        for (int off = 16; off > 0; off >>= 1) {          // LN over 256 ch (1 wave)
            ls += __shfl_xor(ls, off, 32);
            lq += __shfl_xor(lq, off, 32);
        }
        const float m  = ls * (1.0f / (float)COUT);
        const float vr = lq * (1.0f / (float)COUT) - m * m;
        const float rstd = rsqrtf(vr + EPS_);
        const size_t obase = (((size_t)b * T_ + t) * N_ + n) * COUT;
        #pragma unroll
        for (int i = 0; i < 8; ++i) {
            int c = lane + 32 * i;
            float yn = (y[i] - m) * rstd * lg[i] + lb[i];
            float ge = 0.5f * yn * (1.0f + erff(yn * 0.70710678118654752f));
            out[obase + c] = ge;
        }
    }
}

// ---------------------------------------------------------------------------
extern "C" void kernel_launch(void* const* d_in, const int* in_sizes, int n_in,
                              void* d_out, int out_size, void* d_ws, size_t ws_size,
                              hipStream_t stream) {
    (void)in_sizes; (void)n_in; (void)out_size; (void)ws_size;
    const float* x     = (const float*)d_in[0];
    const float* A     = (const float*)d_in[1];
    const float* adj   = (const float*)d_in[2];
    const float* dw    = (const float*)d_in[3];
    const float* Wpw   = (const float*)d_in[4];
    const float* Wconv = (const float*)d_in[5];
    const float* gng   = (const float*)d_in[6];
    const float* gnb   = (const float*)d_in[7];
    const float* lng   = (const float*)d_in[8];
    const float* lnb   = (const float*)d_in[9];
    const float* Wres  = (const float*)d_in[10];
    float* out = (float*)d_out;
    float* ws  = (float*)d_ws;

    float* s_ws = ws;                                    // 256 floats (141 used)
    float* Wcat = ws + 256;                              // 47*512*128
    float* hres = Wcat + (size_t)N_ * OP_ * CIN;         // 8*47*512*512

    pass0_s<<<1, 256, 0, stream>>>(A, adj, s_ws);
    pass0_wcat<<<N_ * OP_, CIN, 0, stream>>>(Wpw, Wres, dw, s_ws, Wcat);
    dim3 g1(T_ / 128, N_, B_);
    pass1_gemm<<<g1, 256, 0, stream>>>(x, Wcat, hres);
    pass2_fuse<<<B_ * N_, 512, 0, stream>>>(hres, Wconv, gng, gnb, lng, lnb, out);
}